// ChebNet_31911607009307
// MI455X (gfx1250) — compile-verified
//
#include <hip/hip_runtime.h>

// ---------------------------------------------------------------------------
// ChebNet on MI455X (gfx1250). fp32 end-to-end; GEMMs on V_WMMA_F32_16X16X4_F32.
// Edge scatter stage is the bottleneck (L2-resident atomics), GEMMs ride free.
// ---------------------------------------------------------------------------

typedef __attribute__((ext_vector_type(2))) float v2f;
typedef __attribute__((ext_vector_type(8))) float v8f;

#define NNODES 100000
#define NEDGES 1600000
#define FDIM   64
#define HDIM   64
#define CDIM   16
#define GDIM   64

// ---- degree: deg[src[e]] += 1 ---------------------------------------------
__global__ void deg_kernel(const int* __restrict__ src, float* __restrict__ deg) {
    int e = blockIdx.x * blockDim.x + threadIdx.x;
    if (e < NEDGES) atomicAdd(&deg[src[e]], 1.0f);
}

// ---- dinv = deg>0 ? rsqrt(deg) : 0 ----------------------------------------
__global__ void dinv_kernel(const float* __restrict__ deg, float* __restrict__ dinv) {
    int i = blockIdx.x * blockDim.x + threadIdx.x;
    if (i < NNODES) {
        float d = deg[i];
        dinv[i] = (d > 0.0f) ? rsqrtf(d) : 0.0f;
    }
}

// ---- norm[e] = dinv[src]*dinv[dst] ----------------------------------------
__global__ void norm_kernel(const int* __restrict__ src, const int* __restrict__ dst,
                            const float* __restrict__ dinv, float* __restrict__ normv) {
    int e = blockIdx.x * blockDim.x + threadIdx.x;
    if (e < NEDGES) normv[e] = dinv[src[e]] * dinv[dst[e]];
}

// ---- tx1[dst] += (-norm[e]) * h[src]  (16 threads/edge, float4 each) ------
__global__ void scatter_kernel(const float* __restrict__ hin, const int* __restrict__ src,
                               const int* __restrict__ dst, const float* __restrict__ normv,
                               float* __restrict__ tx1) {
    int tid = blockIdx.x * blockDim.x + threadIdx.x;
    int e = tid >> 4;
    if (e >= NEDGES) return;
    int c = (tid & 15) << 2;
    int s = src[e], d = dst[e];
    float w = -normv[e];
    const float4 hv = *reinterpret_cast<const float4*>(hin + (size_t)s * HDIM + c);
    float* o = tx1 + (size_t)d * HDIM + c;
    atomicAdd(o + 0, w * hv.x);
    atomicAdd(o + 1, w * hv.y);
    atomicAdd(o + 2, w * hv.z);
    atomicAdd(o + 3, w * hv.w);
}

// ---- WMMA GEMM: out = [relu]( A@W0 [+ Bm@W1] + bias ) ---------------------
// One wave per 16x16 output tile. K multiple of 4; M multiple of 16; Nc
// multiple of 16; weight leading dim == Nc (true for all call sites here).
//
// fp32 WMMA layouts (ISA 7.12.2):
//   A 16x4:  lane m=lane&15; VGPR0/1 hold K = 2*(lane>>4) + {0,1}
//   B 4x16:  lane n=lane&15; VGPR0/1 hold K = 2*(lane>>4) + {0,1}
//   C/D:     8 VGPRs; row = v + 8*(lane>>4), col = lane&15
template <bool RELU, bool HAS_B>
__global__ void wmma_gemm(const float* __restrict__ A, const float* __restrict__ Bm,
                          const float* __restrict__ W0, const float* __restrict__ W1,
                          const float* __restrict__ bias, float* __restrict__ out,
                          int M, int K, int Nc) {
    int wave = (blockIdx.x * blockDim.x + threadIdx.x) >> 5;
    int lane = threadIdx.x & 31;
    int tiles_n = Nc >> 4;
    int tm = wave / tiles_n;
    int tn = wave % tiles_n;
    if (tm * 16 >= M) return;  // wave-uniform: EXEC stays all-1s for WMMA

    const int idx15 = lane & 15;     // A-row / B-col / bias-col for this lane
    const int half  = lane >> 4;     // 0: K+{0,1}, 1: K+{2,3}

    v8f acc = {};
    const float* Arow  = A + (size_t)(tm * 16 + idx15) * K;
    const float* w0col = W0 + tn * 16 + idx15;

    for (int k = 0; k < K; k += 4) {
        int kk = k + half * 2;
        v2f a; a.x = Arow[kk];                 a.y = Arow[kk + 1];
        v2f b; b.x = w0col[(size_t)kk * Nc];   b.y = w0col[(size_t)(kk + 1) * Nc];
        acc = __builtin_amdgcn_wmma_f32_16x16x4_f32(false, a, false, b,
                                                    (short)0, acc, false, false);
    }
    if (HAS_B) {
        const float* Brow  = Bm + (size_t)(tm * 16 + idx15) * K;
        const float* w1col = W1 + tn * 16 + idx15;
        for (int k = 0; k < K; k += 4) {
            int kk = k + half * 2;
            v2f a; a.x = Brow[kk];                 a.y = Brow[kk + 1];
            v2f b; b.x = w1col[(size_t)kk * Nc];   b.y = w1col[(size_t)(kk + 1) * Nc];
            acc = __builtin_amdgcn_wmma_f32_16x16x4_f32(false, a, false, b,
                                                        (short)0, acc, false, false);
        }
    }

    float bv = bias[tn * 16 + idx15];
    #pragma unroll
    for (int v = 0; v < 8; ++v) {
        float r = acc[v] + bv;
        if (RELU) r = fmaxf(r, 0.0f);
        out[(size_t)(tm * 16 + v + half * 8) * Nc + tn * 16 + idx15] = r;
    }
}

// ---- pool: sums[batch[n]] += h[n]; cnt[batch[n]] += 1 ---------------------
__global__ void pool_sum_kernel(const float* __restrict__ h, const int* __restrict__ batch,
                                float* __restrict__ sums, float* __restrict__ cnt) {
    int tid = blockIdx.x * blockDim.x + threadIdx.x;
    int n = tid >> 4;
    if (n >= NNODES) return;
    int c = (tid & 15) << 2;
    int g = batch[n];
    const float4 hv = *reinterpret_cast<const float4*>(h + (size_t)n * HDIM + c);
    float* o = sums + (size_t)g * HDIM + c;
    atomicAdd(o + 0, hv.x);
    atomicAdd(o + 1, hv.y);
    atomicAdd(o + 2, hv.z);
    atomicAdd(o + 3, hv.w);
    if (c == 0) atomicAdd(&cnt[g], 1.0f);
}

__global__ void pool_div_kernel(const float* __restrict__ sums, const float* __restrict__ cnt,
                                float* __restrict__ pooled) {
    int i = blockIdx.x * blockDim.x + threadIdx.x;
    if (i < GDIM * HDIM) pooled[i] = sums[i] / fmaxf(cnt[i >> 6], 1.0f);
}

// ---------------------------------------------------------------------------
extern "C" void kernel_launch(void* const* d_in, const int* in_sizes, int n_in,
                              void* d_out, int out_size, void* d_ws, size_t ws_size,
                              hipStream_t stream) {
    const float* x     = (const float*)d_in[0];
    const int*   edge  = (const int*)d_in[1];   // [2,E] row-major
    const int*   src   = edge;
    const int*   dst   = edge + NEDGES;
    const int*   batch = (const int*)d_in[2];
    const float* W1    = (const float*)d_in[3];
    const float* b1    = (const float*)d_in[4];
    const float* W2    = (const float*)d_in[5];
    const float* b2    = (const float*)d_in[6];
    const float* W3    = (const float*)d_in[7];
    const float* b3    = (const float*)d_in[8];
    const float* Wlin  = (const float*)d_in[9];
    const float* blin  = (const float*)d_in[10];
    float* out = (float*)d_out;

    // workspace layout (floats): ~84 MB total
    float* deg    = (float*)d_ws;                       // N
    float* dinv   = deg    + NNODES;                    // N
    float* normv  = dinv   + NNODES;                    // E
    float* tx1    = normv  + NEDGES;                    // N*H
    float* hA     = tx1    + (size_t)NNODES * HDIM;     // N*H
    float* hB     = hA     + (size_t)NNODES * HDIM;     // N*H
    float* sums   = hB     + (size_t)NNODES * HDIM;     // G*H
    float* cnt    = sums   + GDIM * HDIM;               // G
    float* pooled = cnt    + GDIM;                      // G*H

    const int T = 256;
    const int eBlocks  = (NEDGES + T - 1) / T;
    const int nBlocks  = (NNODES + T - 1) / T;
    const int sBlocks  = (NEDGES * 16 + T - 1) / T;     // scatter: 16 thr/edge
    const int pBlocks  = (NNODES * 16 + T - 1) / T;     // pool: 16 thr/node
    const int gemmBlocks = (NNODES / 16) * (HDIM / 16) / 4;  // 4 waves/block

    // normalization
    hipMemsetAsync(deg, 0, NNODES * sizeof(float), stream);
    deg_kernel <<<eBlocks, T, 0, stream>>>(src, deg);
    dinv_kernel<<<nBlocks, T, 0, stream>>>(deg, dinv);
    norm_kernel<<<eBlocks, T, 0, stream>>>(src, dst, dinv, normv);

    // layer 1: x -> hA (relu)
    hipMemsetAsync(tx1, 0, (size_t)NNODES * HDIM * sizeof(float), stream);
    scatter_kernel<<<sBlocks, T, 0, stream>>>(x, src, dst, normv, tx1);
    wmma_gemm<true, true><<<gemmBlocks, 128, 0, stream>>>(
        x, tx1, W1, W1 + FDIM * HDIM, b1, hA, NNODES, FDIM, HDIM);

    // layer 2: hA -> hB (relu)
    hipMemsetAsync(tx1, 0, (size_t)NNODES * HDIM * sizeof(float), stream);
    scatter_kernel<<<sBlocks, T, 0, stream>>>(hA, src, dst, normv, tx1);
    wmma_gemm<true, true><<<gemmBlocks, 128, 0, stream>>>(
        hA, tx1, W2, W2 + HDIM * HDIM, b2, hB, NNODES, HDIM, HDIM);

    // layer 3: hB -> hA (no relu)
    hipMemsetAsync(tx1, 0, (size_t)NNODES * HDIM * sizeof(float), stream);
    scatter_kernel<<<sBlocks, T, 0, stream>>>(hB, src, dst, normv, tx1);
    wmma_gemm<false, true><<<gemmBlocks, 128, 0, stream>>>(
        hB, tx1, W3, W3 + HDIM * HDIM, b3, hA, NNODES, HDIM, HDIM);

    // global mean pool
    hipMemsetAsync(sums, 0, (GDIM * HDIM + GDIM) * sizeof(float), stream); // sums+cnt
    pool_sum_kernel<<<pBlocks, T, 0, stream>>>(hA, batch, sums, cnt);
    pool_div_kernel<<<(GDIM * HDIM + T - 1) / T, T, 0, stream>>>(sums, cnt, pooled);

    // head: pooled[64x64] @ Wlin[64x16] + blin -> out[64x16]  (4 tiles, 1 block)
    wmma_gemm<false, false><<<1, 128, 0, stream>>>(
        pooled, nullptr, Wlin, nullptr, blin, out, GDIM, HDIM, CDIM);
}